// soft_dice_cldice_72533407695415
// MI455X (gfx1250) — compile-verified
//
#include <hip/hip_runtime.h>
#include <math.h>

// ---------------------------------------------------------------------------
// clDice loss for [4,1,128,128,128] f32 volumes on gfx1250 (MI455X).
//   W=128, H=128, D=128; float4 vectorized: 32 f4/row, HSTRIDE4=32, DSTRIDE4=4096
// ---------------------------------------------------------------------------

#define WMASK 127
#define HSHIFT4 5
#define DSHIFT4 12
#define W4MASK 31
#define HSTRIDE4 32
#define DSTRIDE4 4096

#define TILE_ELEMS 2048           // 16 rows of 128 floats; row-aligned (W halo-free)
#define RED_GRID 1024
#define RED_BLOCK 256

typedef __attribute__((ext_vector_type(2)))  float        v2f;
typedef __attribute__((ext_vector_type(8)))  float        v8f;
typedef __attribute__((ext_vector_type(4)))  unsigned int u32x4;
typedef __attribute__((ext_vector_type(4)))  int          i32x4;
typedef __attribute__((ext_vector_type(8)))  int          i32x8;

// ------------------------------- device helpers ----------------------------

__device__ __forceinline__ float sigf(float x) {
  return 1.0f / (1.0f + expf(-x));
}

__device__ __forceinline__ float4 min4(float4 a, float4 b) {
  return make_float4(fminf(a.x, b.x), fminf(a.y, b.y), fminf(a.z, b.z), fminf(a.w, b.w));
}
__device__ __forceinline__ float4 max4(float4 a, float4 b) {
  return make_float4(fmaxf(a.x, b.x), fmaxf(a.y, b.y), fmaxf(a.z, b.z), fmaxf(a.w, b.w));
}

// WMMA-based accumulation: D = A x ones + C. With B == ones, every column of D
// holds the row sums of A, so sum(all of D) == 16 * sum(all of A).
__device__ __forceinline__ v8f wacc(v8f c, float a0, float a1) {
  v2f a; a.x = a0; a.y = a1;
  v2f b; b.x = 1.0f; b.y = 1.0f;
  return __builtin_amdgcn_wmma_f32_16x16x4_f32(
      /*neg_a=*/false, a, /*neg_b=*/false, b,
      /*c_mod=*/(short)0, c, /*reuse_a=*/false, /*reuse_b=*/false);
}

__device__ __forceinline__ float vsum8(v8f c) {
  float s = 0.0f;
#pragma unroll
  for (int k = 0; k < 8; ++k) s += c[k];
  return s;
}

__device__ __forceinline__ float wave_sum(float v) {
#pragma unroll
  for (int off = 16; off > 0; off >>= 1) v += __shfl_down(v, off, 32);
  return v;
}

// ------------------------------- elementwise -------------------------------

__global__ void sigmoid_k(const float4* __restrict__ x, float4* __restrict__ y, int n4) {
  int i = blockIdx.x * blockDim.x + threadIdx.x;
  int stride = gridDim.x * blockDim.x;
  for (; i < n4; i += stride) {
    float4 v = x[i];
    y[i] = make_float4(sigf(v.x), sigf(v.y), sigf(v.z), sigf(v.w));
  }
}

__global__ void copy_k(const float4* __restrict__ x, float4* __restrict__ y, int n4) {
  int i = blockIdx.x * blockDim.x + threadIdx.x;
  int stride = gridDim.x * blockDim.x;
  for (; i < n4; i += stride) y[i] = x[i];
}

__global__ void zero_acc_k(float* acc) {
  if (threadIdx.x < 8) acc[threadIdx.x] = 0.0f;
}

// ------------------------------- stencils ----------------------------------
// Lanes map to consecutive float4s: 32 lanes == one 128-wide row, so the h/d
// boundary branches below are wave-uniform (no divergence); only the W-shift
// is per-lane, handled with +/-INF sentinels (min/max identity).

// soft_erode: min over the 7-point cross (SAME padding == ignore OOB taps).
__global__ void erode_k(const float* __restrict__ s, float4* __restrict__ dst, int n4) {
  const float4* s4 = (const float4*)s;
  int i = blockIdx.x * blockDim.x + threadIdx.x;
  int stride = gridDim.x * blockDim.x;
  for (; i < n4; i += stride) {
    int w4 = i & W4MASK;
    int h = (i >> HSHIFT4) & WMASK;
    int d = (i >> DSHIFT4) & WMASK;
    float4 c = s4[i];
    const float* sc = s + 4 * i;
    float left  = (w4 > 0)      ? sc[-1] : __builtin_inff();
    float right = (w4 < W4MASK) ? sc[4]  : __builtin_inff();
    float4 m;   // min over {center, w-1, w+1}
    m.x = fminf(fminf(left, c.x), c.y);
    m.y = fminf(fminf(c.x, c.y), c.z);
    m.z = fminf(fminf(c.y, c.z), c.w);
    m.w = fminf(fminf(c.z, c.w), right);
    if (h > 0)     m = min4(m, s4[i - HSTRIDE4]);
    if (h < 127)   m = min4(m, s4[i + HSTRIDE4]);
    if (d > 0)     m = min4(m, s4[i - DSTRIDE4]);
    if (d < 127)   m = min4(m, s4[i + DSTRIDE4]);
    dst[i] = m;
  }
}

// Max over W-window of 3. Tile of 16 full rows staged into LDS via the
// Tensor Data Mover (gfx1250 TDM, TENSORcnt). Tiles are row-aligned, so no
// cross-tile halo is needed for the W direction.
__global__ void dilw_k(const float* __restrict__ src, float4* __restrict__ dst) {
  __shared__ __align__(16) float tile[TILE_ELEMS];
  const unsigned base = blockIdx.x * (unsigned)TILE_ELEMS;

  if (threadIdx.x < 32u) {  // one wave issues the DMA (EXEC ignored by TDM)
    unsigned lds = (unsigned)(size_t)&tile[0];
    unsigned long long ga = (unsigned long long)(size_t)(src + base);
    u32x4 g0;
    g0.x = 1u;                                     // count=1, user descriptor
    g0.y = lds;                                    // lds_addr (bytes)
    g0.z = (unsigned)ga;                           // global_addr[31:0]
    g0.w = (unsigned)((ga >> 32) & 0x01FFFFFFull)  // global_addr[56:32]
           | 0x80000000u;                          // type=2 ("image")
    i32x8 g1;
    g1[0] = (int)(2u << 16);                       // data_size=4B, mask=0
    g1[1] = (int)((TILE_ELEMS & 0xFFFFu) << 16);   // tensor_dim0[15:0]
    g1[2] = (int)(1u << 16);                       // tensor_dim0 hi=0, tensor_dim1=1
    g1[3] = (int)((TILE_ELEMS & 0xFFFFu) << 16);   // tile_dim0 = 2048
    g1[4] = 0;                                     // tile_dim1/2 unused
    g1[5] = TILE_ELEMS;                            // tensor_dim0_stride
    g1[6] = 0;
    g1[7] = 0;
    i32x4 z4 = {0, 0, 0, 0};
#if __clang_major__ >= 23
    i32x8 z8 = {0, 0, 0, 0, 0, 0, 0, 0};
    __builtin_amdgcn_tensor_load_to_lds(g0, g1, z4, z4, z8, 0);
#else
    __builtin_amdgcn_tensor_load_to_lds(g0, g1, z4, z4, 0);
#endif
    __builtin_amdgcn_s_wait_tensorcnt(0);
  }
  __syncthreads();

  const float4* tile4 = (const float4*)tile;
#pragma unroll
  for (int k = 0; k < TILE_ELEMS / (256 * 4); ++k) {
    unsigned q = threadIdx.x + k * 256u;           // f4 index in tile, 0..511
    unsigned w4 = q & W4MASK;
    float4 c = tile4[q];
    float left  = (w4 > 0)      ? tile[q * 4 - 1] : -__builtin_inff();
    float right = (w4 < W4MASK) ? tile[q * 4 + 4] : -__builtin_inff();
    float4 m;
    m.x = fmaxf(fmaxf(left, c.x), c.y);
    m.y = fmaxf(fmaxf(c.x, c.y), c.z);
    m.z = fmaxf(fmaxf(c.y, c.z), c.w);
    m.w = fmaxf(fmaxf(c.z, c.w), right);
    dst[base / 4 + q] = m;
  }
}

__global__ void dilh_k(const float4* __restrict__ s, float4* __restrict__ dst, int n4) {
  int i = blockIdx.x * blockDim.x + threadIdx.x;
  int stride = gridDim.x * blockDim.x;
  for (; i < n4; i += stride) {
    int h = (i >> HSHIFT4) & WMASK;
    float4 m = s[i];
    if (h > 0)   m = max4(m, s[i - HSTRIDE4]);
    if (h < 127) m = max4(m, s[i + HSTRIDE4]);
    dst[i] = m;
  }
}

// Final dilate direction (D) fused with skeleton init: skel = relu(x - open(x))
__global__ void dild_init_k(const float4* __restrict__ s, const float4* __restrict__ x,
                            float4* __restrict__ skel, int n4) {
  int i = blockIdx.x * blockDim.x + threadIdx.x;
  int stride = gridDim.x * blockDim.x;
  for (; i < n4; i += stride) {
    int d = (i >> DSHIFT4) & WMASK;
    float4 o = s[i];
    if (d > 0)   o = max4(o, s[i - DSTRIDE4]);
    if (d < 127) o = max4(o, s[i + DSTRIDE4]);
    float4 xv = x[i];
    skel[i] = make_float4(fmaxf(xv.x - o.x, 0.0f), fmaxf(xv.y - o.y, 0.0f),
                          fmaxf(xv.z - o.z, 0.0f), fmaxf(xv.w - o.w, 0.0f));
  }
}

// Final dilate direction (D) fused with skeleton update:
// delta = relu(img - open); skel += relu(delta - skel*delta)
__global__ void dild_upd_k(const float4* __restrict__ s, const float4* __restrict__ img,
                           float4* __restrict__ skel, int n4) {
  int i = blockIdx.x * blockDim.x + threadIdx.x;
  int stride = gridDim.x * blockDim.x;
  for (; i < n4; i += stride) {
    int d = (i >> DSHIFT4) & WMASK;
    float4 o = s[i];
    if (d > 0)   o = max4(o, s[i - DSTRIDE4]);
    if (d < 127) o = max4(o, s[i + DSTRIDE4]);
    float4 im = img[i];
    float4 sk = skel[i];
    float dx = fmaxf(im.x - o.x, 0.0f);
    float dy = fmaxf(im.y - o.y, 0.0f);
    float dz = fmaxf(im.z - o.z, 0.0f);
    float dw = fmaxf(im.w - o.w, 0.0f);
    skel[i] = make_float4(sk.x + fmaxf(dx - sk.x * dx, 0.0f),
                          sk.y + fmaxf(dy - sk.y * dy, 0.0f),
                          sk.z + fmaxf(dz - sk.z * dz, 0.0f),
                          sk.w + fmaxf(dw - sk.w * dw, 0.0f));
  }
}

// ------------------------------- reductions (WMMA) -------------------------

// acc[0]+=sum(y_true), acc[1]+=sum(sigmoid(pred)), acc[2]+=sum(y_true*sigmoid(pred))
__global__ void reduce_dice_k(const float* __restrict__ pred,
                              const float* __restrict__ yt,
                              float* __restrict__ acc, int n, int iters) {
  const unsigned lanes = gridDim.x * blockDim.x;
  const unsigned tid = blockIdx.x * blockDim.x + threadIdx.x;
  v8f c0 = {}, c1 = {}, c2 = {};
  for (int it = 0; it < iters; ++it) {  // uniform trip count: EXEC all-ones
    unsigned i = ((unsigned)it * lanes + tid) * 2u;
    float x0 = pred[i], x1 = pred[i + 1];
    float p0 = sigf(x0), p1 = sigf(x1);
    float t0 = yt[i], t1 = yt[i + 1];
    c0 = wacc(c0, t0, t1);
    c1 = wacc(c1, p0, p1);
    c2 = wacc(c2, t0 * p0, t1 * p1);
  }
  float s0 = vsum8(c0), s1 = vsum8(c1), s2 = vsum8(c2);
  // scalar tail (in case n not divisible; divergence only after all WMMA)
  for (unsigned i = (unsigned)iters * lanes * 2u + tid; i < (unsigned)n; i += lanes) {
    float p = sigf(pred[i]);
    float t = yt[i];
    s0 += 16.0f * t; s1 += 16.0f * p; s2 += 16.0f * t * p;
  }
  s0 = wave_sum(s0); s1 = wave_sum(s1); s2 = wave_sum(s2);
  if ((threadIdx.x & 31u) == 0u) {
    atomicAdd(&acc[0], s0 * (1.0f / 16.0f));
    atomicAdd(&acc[1], s1 * (1.0f / 16.0f));
    atomicAdd(&acc[2], s2 * (1.0f / 16.0f));
  }
}

// accA += sum(skel), accB += sum(skel * f(other)), f = sigmoid if sig_other
__global__ void reduce_skel_k(const float* __restrict__ skel,
                              const float* __restrict__ other,
                              float* __restrict__ accA, float* __restrict__ accB,
                              int n, int iters, int sig_other) {
  const unsigned lanes = gridDim.x * blockDim.x;
  const unsigned tid = blockIdx.x * blockDim.x + threadIdx.x;
  v8f c0 = {}, c1 = {};
  for (int it = 0; it < iters; ++it) {
    unsigned i = ((unsigned)it * lanes + tid) * 2u;
    float s0v = skel[i], s1v = skel[i + 1];
    float o0 = other[i], o1 = other[i + 1];
    if (sig_other) { o0 = sigf(o0); o1 = sigf(o1); }  // uniform branch
    c0 = wacc(c0, s0v, s1v);
    c1 = wacc(c1, s0v * o0, s1v * o1);
  }
  float s0 = vsum8(c0), s1 = vsum8(c1);
  for (unsigned i = (unsigned)iters * lanes * 2u + tid; i < (unsigned)n; i += lanes) {
    float sv = skel[i];
    float ov = other[i];
    if (sig_other) ov = sigf(ov);
    s0 += 16.0f * sv; s1 += 16.0f * sv * ov;
  }
  s0 = wave_sum(s0); s1 = wave_sum(s1);
  if ((threadIdx.x & 31u) == 0u) {
    atomicAdd(accA, s0 * (1.0f / 16.0f));
    atomicAdd(accB, s1 * (1.0f / 16.0f));
  }
}

__global__ void finalize_k(const float* __restrict__ acc, float* __restrict__ out) {
  if (threadIdx.x == 0 && blockIdx.x == 0) {
    const float SM = 1.0f;
    float dice = 1.0f - (2.0f * acc[2] + SM) / (acc[0] + acc[1] + SM);
    float tprec = (acc[4] + SM) / (acc[3] + SM);
    float tsens = (acc[6] + SM) / (acc[5] + SM);
    float cl = 1.0f - 2.0f * (tprec * tsens) / (tprec + tsens);
    out[0] = 0.7f * dice + 0.3f * cl;
  }
}

// ------------------------------- host orchestration ------------------------

extern "C" void kernel_launch(void* const* d_in, const int* in_sizes, int n_in,
                              void* d_out, int out_size, void* d_ws, size_t ws_size,
                              hipStream_t stream) {
  (void)n_in; (void)out_size; (void)ws_size;
  const float* pred = (const float*)d_in[0];
  const float* yt   = (const float*)d_in[1];
  float* out = (float*)d_out;
  const int n = in_sizes[0];  // 8,388,608 for (4,1,128,128,128)
  const int n4 = n >> 2;

  float* imgA = (float*)d_ws;
  float* imgB = imgA + n;
  float* t0   = imgB + n;
  float* skel = t0 + n;
  float* acc  = skel + n;     // 8 floats

  const dim3 blk(256);
  const dim3 grd(2048);                       // grid-stride, float4 kernels
  const dim3 grd_w(n / TILE_ELEMS);           // 4096 tiles of 2048 (row-aligned)
  const int red_iters = n / (2 * RED_GRID * RED_BLOCK);  // 16

  zero_acc_k<<<1, 32, 0, stream>>>(acc);
  reduce_dice_k<<<RED_GRID, RED_BLOCK, 0, stream>>>(pred, yt, acc, n, red_iters);

  for (int field = 0; field < 2; ++field) {
    if (field == 0)
      sigmoid_k<<<grd, blk, 0, stream>>>((const float4*)pred, (float4*)imgA, n4);
    else
      copy_k<<<grd, blk, 0, stream>>>((const float4*)yt, (float4*)imgA, n4);

    float* cur = imgA;
    float* oth = imgB;

    // skel = relu(x - open(x)):  open = dilD(dilH(dilW(erode(x))))
    erode_k<<<grd, blk, 0, stream>>>(cur, (float4*)oth, n4);
    dilw_k<<<grd_w, blk, 0, stream>>>(oth, (float4*)t0);
    dilh_k<<<grd, blk, 0, stream>>>((const float4*)t0, (float4*)oth, n4);
    dild_init_k<<<grd, blk, 0, stream>>>((const float4*)oth, (const float4*)cur,
                                         (float4*)skel, n4);

    for (int it = 0; it < 16; ++it) {
      erode_k<<<grd, blk, 0, stream>>>(cur, (float4*)oth, n4);   // img = erode(img)
      erode_k<<<grd, blk, 0, stream>>>(oth, (float4*)t0, n4);    // erode(img)
      dilw_k<<<grd_w, blk, 0, stream>>>(t0, (float4*)cur);       // cur free -> reuse
      dilh_k<<<grd, blk, 0, stream>>>((const float4*)cur, (float4*)t0, n4);
      dild_upd_k<<<grd, blk, 0, stream>>>((const float4*)t0, (const float4*)oth,
                                          (float4*)skel, n4);    // fused update
      float* tmp = cur; cur = oth; oth = tmp;                    // img now in cur
    }

    if (field == 0)
      reduce_skel_k<<<RED_GRID, RED_BLOCK, 0, stream>>>(skel, yt, &acc[3], &acc[4],
                                                        n, red_iters, 0);
    else
      reduce_skel_k<<<RED_GRID, RED_BLOCK, 0, stream>>>(skel, pred, &acc[5], &acc[6],
                                                        n, red_iters, 1);
  }

  finalize_k<<<1, 32, 0, stream>>>(acc, out);
}